// EndEffectorLoss_90039694393961
// MI455X (gfx1250) — compile-verified
//
#include <hip/hip_runtime.h>
#include <stdint.h>

typedef __attribute__((ext_vector_type(2))) float v2f;
typedef __attribute__((ext_vector_type(8))) float v8f;

#define NJ 24
#define FPB 128              // frames per block (4 waves)
#define POSE_STRIDE_W 148    // dwords/frame in LDS: 144 data + 4 pad (16B-aligned stride, 2-way banks)
#define OFF_STRIDE_W  76     // dwords/frame in LDS: 72 data + 4 pad

// ---------- LDS raw-offset + async-DMA helpers (CDNA5) ----------
typedef const __attribute__((address_space(3))) void lds_cv_t;

__device__ __forceinline__ uint32_t lds_raw(const void* p) {
    // addrspacecast generic -> LDS, then ptrtoint gives the raw LDS byte offset
    return (uint32_t)(uintptr_t)(lds_cv_t*)p;
}

// global_load_async_to_lds_b128: VDST = LDS byte address (VGPR),
// VADDR = 32-bit byte offset (VGPR), SADDR = 64-bit base (SGPR pair). GVS mode.
__device__ __forceinline__ void async_b128(uint32_t lds_addr, const float* gbase, uint32_t goff) {
    asm volatile("global_load_async_to_lds_b128 %0, %1, %2"
                 :: "v"(lds_addr), "v"(goff), "s"(gbase) : "memory");
}

__device__ __forceinline__ void wait_async0() {
    asm volatile("s_wait_asynccnt 0" ::: "memory");
}

// Stage one skeleton's block slab (FPB frames) into LDS, coalesced 16B chunks.
// pose: FPB*576B = FPB*36 chunks; off: FPB*288B = FPB*18 chunks.
__device__ __forceinline__ void stage_async(uint32_t poseBase, uint32_t offBase,
                                            const float* __restrict__ poseG,
                                            const float* __restrict__ offG, uint32_t t) {
#pragma unroll
    for (uint32_t i = 0; i < 36; ++i) {
        uint32_t n = t + i * FPB;              // chunk id, lane-contiguous -> coalesced global
        uint32_t fr = n / 36u;                 // frame
        uint32_t within = (n - fr * 36u) * 16u;
        async_b128(poseBase + fr * (POSE_STRIDE_W * 4) + within, poseG, n * 16u);
    }
#pragma unroll
    for (uint32_t i = 0; i < 18; ++i) {
        uint32_t n = t + i * FPB;
        uint32_t fr = n / 18u;
        uint32_t within = (n - fr * 18u) * 16u;
        async_b128(offBase + fr * (OFF_STRIDE_W * 4) + within, offG, n * 16u);
    }
}

// ---------- small 3x3 helpers (row-major) ----------
__device__ __forceinline__ void rot6(const float* __restrict__ p, float m[9]) {
    float x0 = p[0], x1 = p[1], x2 = p[2];
    float y0 = p[3], y1 = p[4], y2 = p[5];
    float inv1 = __frsqrt_rn(x0 * x0 + x1 * x1 + x2 * x2);
    float b10 = x0 * inv1, b11 = x1 * inv1, b12 = x2 * inv1;
    float d = b10 * y0 + b11 * y1 + b12 * y2;
    float a0 = y0 - d * b10, a1 = y1 - d * b11, a2 = y2 - d * b12;
    float inv2 = __frsqrt_rn(a0 * a0 + a1 * a1 + a2 * a2);
    float b20 = a0 * inv2, b21 = a1 * inv2, b22 = a2 * inv2;
    m[0] = b10; m[1] = b11; m[2] = b12;
    m[3] = b20; m[4] = b21; m[5] = b22;
    m[6] = b11 * b22 - b12 * b21;
    m[7] = b12 * b20 - b10 * b22;
    m[8] = b10 * b21 - b11 * b20;
}

__device__ __forceinline__ void mm3(const float A[9], const float B[9], float C[9]) {
#pragma unroll
    for (int i = 0; i < 3; ++i)
#pragma unroll
        for (int j = 0; j < 3; ++j)
            C[3 * i + j] = A[3 * i + 0] * B[0 + j] + A[3 * i + 1] * B[3 + j] + A[3 * i + 2] * B[6 + j];
}

__device__ __forceinline__ void stepJ(const float* __restrict__ pose, const float* __restrict__ off,
                                      int j, float O[9], float p[3]) {
    float ox = off[3 * j + 0], oy = off[3 * j + 1], oz = off[3 * j + 2];
    p[0] += O[0] * ox + O[1] * oy + O[2] * oz;
    p[1] += O[3] * ox + O[4] * oy + O[5] * oz;
    p[2] += O[6] * ox + O[7] * oy + O[8] * oz;
    float R[9]; rot6(pose + 6 * j, R);
    float T[9]; mm3(O, R, T);
#pragma unroll
    for (int k = 0; k < 9; ++k) O[k] = T[k];
}

__device__ __forceinline__ void stepEnd(const float* __restrict__ off, int j,
                                        const float O[9], const float p[3], float out[3]) {
    float ox = off[3 * j + 0], oy = off[3 * j + 1], oz = off[3 * j + 2];
    out[0] = p[0] + O[0] * ox + O[1] * oy + O[2] * oz;
    out[1] = p[1] + O[3] * ox + O[4] * oy + O[5] * oz;
    out[2] = p[2] + O[6] * ox + O[7] * oy + O[8] * oz;
}

// DFS over fixed tree; branch points at joints 0 and 9. ee order = [10,11,15,22,23].
__device__ __forceinline__ void fk5(const float* __restrict__ pose, const float* __restrict__ off,
                                    float ee[5][3]) {
    float O[9], p[3];
    rot6(pose, O);
    p[0] = off[0]; p[1] = off[1]; p[2] = off[2];

    float O0[9], p0[3];
#pragma unroll
    for (int k = 0; k < 9; ++k) O0[k] = O[k];
    p0[0] = p[0]; p0[1] = p[1]; p0[2] = p[2];

    stepJ(pose, off, 1, O, p); stepJ(pose, off, 4, O, p); stepJ(pose, off, 7, O, p);
    stepEnd(off, 10, O, p, ee[0]);

#pragma unroll
    for (int k = 0; k < 9; ++k) O[k] = O0[k];
    p[0] = p0[0]; p[1] = p0[1]; p[2] = p0[2];
    stepJ(pose, off, 2, O, p); stepJ(pose, off, 5, O, p); stepJ(pose, off, 8, O, p);
    stepEnd(off, 11, O, p, ee[1]);

#pragma unroll
    for (int k = 0; k < 9; ++k) O[k] = O0[k];
    p[0] = p0[0]; p[1] = p0[1]; p[2] = p0[2];
    stepJ(pose, off, 3, O, p); stepJ(pose, off, 6, O, p); stepJ(pose, off, 9, O, p);

    float O9[9], p9[3];
#pragma unroll
    for (int k = 0; k < 9; ++k) O9[k] = O[k];
    p9[0] = p[0]; p9[1] = p[1]; p9[2] = p[2];

    stepJ(pose, off, 12, O, p);
    stepEnd(off, 15, O, p, ee[2]);

#pragma unroll
    for (int k = 0; k < 9; ++k) O[k] = O9[k];
    p[0] = p9[0]; p[1] = p9[1]; p[2] = p9[2];
    stepJ(pose, off, 13, O, p); stepJ(pose, off, 16, O, p);
    stepJ(pose, off, 18, O, p); stepJ(pose, off, 20, O, p);
    stepEnd(off, 22, O, p, ee[3]);

#pragma unroll
    for (int k = 0; k < 9; ++k) O[k] = O9[k];
    p[0] = p9[0]; p[1] = p9[1]; p[2] = p9[2];
    stepJ(pose, off, 14, O, p); stepJ(pose, off, 17, O, p);
    stepJ(pose, off, 19, O, p); stepJ(pose, off, 21, O, p);
    stepEnd(off, 23, O, p, ee[4]);
}

// ---------- CDNA5 tensor-core wave reduction ----------
// V_WMMA_F32_16X16X4_F32, B = ones: D[m,n] = A[m,0..3] row-sum.
// Lane l contributes A[l%16, l<16?0:2]; rowsum(m) = v[m] + v[m+16].
// D lanes<16 hold rows 0..7, lanes>=16 rows 8..15 -> sum 8 regs, then swap halves.
__device__ __forceinline__ float wave_reduce_wmma(float v) {
    v2f a; a[0] = v;    a[1] = 0.0f;
    v2f b; b[0] = 1.0f; b[1] = 1.0f;
    v8f c = {};
    v8f d = __builtin_amdgcn_wmma_f32_16x16x4_f32(false, a, false, b, (short)0, c, false, false);
    float s = ((d[0] + d[1]) + (d[2] + d[3])) + ((d[4] + d[5]) + (d[6] + d[7]));
    // SWAPX16 via ds_swizzle: offset = xor(0x10)<<10 | and(0x1f) = 0x401F
    s += __int_as_float(__builtin_amdgcn_ds_swizzle(__float_as_int(s), 0x401F));
    return s;   // full 32-lane sum on every lane
}

__device__ __forceinline__ float block_reduce(float v, float* lds) {
    float s = wave_reduce_wmma(v);
    int lane = threadIdx.x & 31;
    int wave = threadIdx.x >> 5;
    if (lane == 0) lds[wave] = s;
    __syncthreads();
    float t = 0.0f;
    if (threadIdx.x == 0) {
        int nw = (blockDim.x + 31) >> 5;
        for (int i = 0; i < nw; ++i) t += lds[i];
    }
    return t;  // valid on thread 0
}

// ---------- setup: t-pose bbox scales + per-end-site constants (uniform) ----------
__global__ void ee_setup(const float* __restrict__ tA, const float* __restrict__ tB,
                         float* __restrict__ cws) {
    if (threadIdx.x != 0 || blockIdx.x != 0) return;
    float mnA[3], mxA[3], mnB[3], mxB[3];
#pragma unroll
    for (int c = 0; c < 3; ++c) { mnA[c] = mxA[c] = tA[c]; mnB[c] = mxB[c] = tB[c]; }
    for (int j = 1; j < NJ; ++j)
#pragma unroll
        for (int c = 0; c < 3; ++c) {
            float a = tA[3 * j + c], b = tB[3 * j + c];
            mnA[c] = fminf(mnA[c], a); mxA[c] = fmaxf(mxA[c], a);
            mnB[c] = fminf(mnB[c], b); mxB[c] = fmaxf(mxB[c], b);
        }
    float invSA[3], invSB[3];
#pragma unroll
    for (int c = 0; c < 3; ++c) {
        invSA[c] = 1.0f / (mxA[c] - mnA[c]);
        invSB[c] = 1.0f / (mxB[c] - mnB[c]);
        cws[c] = invSA[c];
        cws[3 + c] = invSB[c];
    }
    const int EE[5] = { 10, 11, 15, 22, 23 };
#pragma unroll
    for (int e = 0; e < 5; ++e)
#pragma unroll
        for (int c = 0; c < 3; ++c)
            cws[6 + 3 * e + c] = tA[3 * EE[e] + c] * invSA[c] - tB[3 * EE[e] + c] * invSB[c];
}

// ---------- main kernel: async-stage 128 frames to LDS, FK per thread ----------
__global__ void __launch_bounds__(FPB) ee_loss_main(
    const float* __restrict__ poseA, const float* __restrict__ poseB,
    const float* __restrict__ offA,  const float* __restrict__ offB,
    const float* __restrict__ cws,   float* __restrict__ partial, int F)
{
    __shared__ __align__(16) float s_pose[FPB * POSE_STRIDE_W];  // ~74 KB
    __shared__ __align__(16) float s_off [FPB * OFF_STRIDE_W];   // ~38 KB
    __shared__ float s_red[FPB / 32];

    const uint32_t t = threadIdx.x;
    const int blockStart = blockIdx.x * FPB;

    // uniform constants (scalar loads from workspace)
    float invSA[3], invSB[3], cst[5][3];
#pragma unroll
    for (int c = 0; c < 3; ++c) { invSA[c] = cws[c]; invSB[c] = cws[3 + c]; }
#pragma unroll
    for (int e = 0; e < 5; ++e)
#pragma unroll
        for (int c = 0; c < 3; ++c) cst[e][c] = cws[6 + 3 * e + c];

    float loss = 0.0f;
    float eeA[5][3], eeB[5][3];

    if (blockStart + FPB <= F) {   // uniform branch: barriers legal
        const uint32_t poseBase = lds_raw(s_pose);
        const uint32_t offBase  = lds_raw(s_off);

        // ---- phase A: DMA -> LDS, FK ----
        stage_async(poseBase, offBase,
                    poseA + (size_t)blockStart * (NJ * 6),
                    offA  + (size_t)blockStart * (NJ * 3), t);
        wait_async0();
        __syncthreads();
        fk5(s_pose + t * POSE_STRIDE_W, s_off + t * OFF_STRIDE_W, eeA);
        __syncthreads();           // everyone done reading A before overwrite

        // ---- phase B ----
        stage_async(poseBase, offBase,
                    poseB + (size_t)blockStart * (NJ * 6),
                    offB  + (size_t)blockStart * (NJ * 3), t);
        wait_async0();
        __syncthreads();
        fk5(s_pose + t * POSE_STRIDE_W, s_off + t * OFF_STRIDE_W, eeB);

#pragma unroll
        for (int e = 0; e < 5; ++e)
#pragma unroll
            for (int c = 0; c < 3; ++c) {
                float d = eeA[e][c] * invSA[c] - eeB[e][c] * invSB[c] - cst[e][c];
                loss += d * d;
            }
    } else {
        // tail block: direct per-thread loads (rare / absent when F % FPB == 0)
        int f = blockStart + (int)t;
        if (f < F) {
            const float* pA = poseA + (size_t)f * (NJ * 6);
            const float* pB = poseB + (size_t)f * (NJ * 6);
            const float* oA = offA + (size_t)f * (NJ * 3);
            const float* oB = offB + (size_t)f * (NJ * 3);
            __builtin_prefetch(pA, 0, 0);
            __builtin_prefetch(pB, 0, 0);
            fk5(pA, oA, eeA);
            fk5(pB, oB, eeB);
#pragma unroll
            for (int e = 0; e < 5; ++e)
#pragma unroll
                for (int c = 0; c < 3; ++c) {
                    float d = eeA[e][c] * invSA[c] - eeB[e][c] * invSB[c] - cst[e][c];
                    loss += d * d;
                }
        }
    }

    float bsum = block_reduce(loss, s_red);
    if (threadIdx.x == 0) partial[blockIdx.x] = bsum;
}

// ---------- finisher ----------
__global__ void __launch_bounds__(256) ee_loss_finish(
    const float* __restrict__ partial, int nb, float* __restrict__ out, float scale)
{
    __shared__ float lds[8];
    float s = 0.0f;
    for (int i = threadIdx.x; i < nb; i += 256) s += partial[i];
    float tsum = block_reduce(s, lds);
    if (threadIdx.x == 0) out[0] = tsum * scale;
}

extern "C" void kernel_launch(void* const* d_in, const int* in_sizes, int n_in,
                              void* d_out, int out_size, void* d_ws, size_t ws_size,
                              hipStream_t stream) {
    const float* poseA = (const float*)d_in[0];
    const float* poseB = (const float*)d_in[1];
    const float* offA  = (const float*)d_in[2];
    const float* offB  = (const float*)d_in[3];
    const float* tA    = (const float*)d_in[4];
    const float* tB    = (const float*)d_in[5];

    int F = in_sizes[0] / (NJ * 6);
    int nb = (F + FPB - 1) / FPB;
    float* cws = (float*)d_ws;          // 21 uniform constants
    float* partial = cws + 64;          // per-block partial sums
    float scale = 1.0f / ((float)F * 15.0f);

    ee_setup<<<1, 32, 0, stream>>>(tA, tB, cws);
    ee_loss_main<<<nb, FPB, 0, stream>>>(poseA, poseB, offA, offB, cws, partial, F);
    ee_loss_finish<<<1, 256, 0, stream>>>(partial, nb, (float*)d_out, scale);
}